// LoOP_48000554500947
// MI455X (gfx1250) — compile-verified
//
#include <hip/hip_runtime.h>
#include <math.h>

// ---- problem constants (from reference) ----
#define N_TRAIN 10000
#define DIMS    512
#define KNN     16
#define BATCH   512
#define NTILES  (N_TRAIN / 16)    // 625 exact
#define LAMBDA  3.0f

// ---- kernel-config constants ----
#define NWAVES  2                  // waves per kNN block
#define BLOCKT  (NWAVES * 32)      // 64 threads
#define QSTRIDE 516                // query row stride (516 % 64 == 4 -> conflict-free)
#define CANDS   (NWAVES * 2 * KNN) // 64 merge candidates per query
#define CSTRIDE (CANDS + 1)        // 65 (conflict-free column scans)
#define BIGF    3.0e38f

// point-tile slice staged by the TDM: 16 rows x 64 f32, padded to stride 68
#define SLICE_COLS   64
#define SLICE_STRIDE 68            // 64 data + 4 pad dwords (68 % 64 == 4)
#define SLICE_FLOATS (16 * SLICE_STRIDE)
#define SLICES       (DIMS / SLICE_COLS)   // 8 slices per 16x512 tile

typedef __attribute__((ext_vector_type(2))) float        v2f;
typedef __attribute__((ext_vector_type(8))) float        v8f;
typedef __attribute__((ext_vector_type(4))) unsigned int v4u;
typedef __attribute__((ext_vector_type(8))) int          v8i;
typedef __attribute__((ext_vector_type(4))) int          v4i;

#if defined(__has_builtin)
#if __has_builtin(__builtin_amdgcn_tensor_load_to_lds) && \
    __has_builtin(__builtin_amdgcn_s_wait_tensorcnt)
#define USE_TDM 1
#endif
#endif

#define SCHED_FENCE() asm volatile("" ::: "memory")

#ifdef USE_TDM
// TDM 2D tile load: 16 rows x 64 f32 from a row-major [*, 512] matrix into LDS,
// LDS-padded +4 dwords per 64 dwords (row stride 68).  D# packing per CDNA5 ISA §8.
__device__ __forceinline__ void tdm_load_slice(const float* gsrc, unsigned int lds_off) {
  const unsigned long long ga = (unsigned long long)(uintptr_t)gsrc;
  v4u g0;
  g0.x = 1u;                                         // count = 1 (valid descriptor)
  g0.y = lds_off;                                    // LDS byte address of tile dest
  g0.z = (unsigned int)(ga & 0xFFFFFFFFull);         // global_addr[31:0]
  g0.w = (unsigned int)((ga >> 32) & 0x1FFFFFFull)   // global_addr[56:32]
       | (2u << 30);                                 // type = 2 ("image")
  v8i g1;
  g1[0] = (int)((2u << 16)        // data_size = 2 -> 4 bytes
              | (1u << 20)        // pad_enable
              | (5u << 22)        // pad_interval code 5 -> every 64 dwords
              | (3u << 25));      // pad_amount  code 3 -> 4 dwords
  g1[1] = (int)(512u << 16);      // tensor_dim0 = 512 (bits 79:48, low half)
  g1[2] = (int)(16u << 16);       // tensor_dim0 hi = 0 ; tensor_dim1 = 16
  g1[3] = (int)((unsigned)SLICE_COLS << 16);  // tensor_dim1 hi = 0 ; tile_dim0 = 64
  g1[4] = 16;                     // tile_dim1 = 16 rows ; tile_dim2 = 0 (2D)
  g1[5] = DIMS;                   // tensor_dim0_stride = 512 (row stride)
  g1[6] = 0;                      // stride hi ; tensor_dim1_stride lo (unused, 2D)
  g1[7] = 0;
  const v4i z4 = {0, 0, 0, 0};
#if defined(__clang_major__) && (__clang_major__ >= 23)
  const v8i z8 = {0, 0, 0, 0, 0, 0, 0, 0};
  __builtin_amdgcn_tensor_load_to_lds(g0, g1, z4, z4, z8, 0);
#else
  __builtin_amdgcn_tensor_load_to_lds(g0, g1, z4, z4, 0);
#endif
}
#else
// fallback: wave-cooperative synchronous copy (same padded layout)
__device__ __forceinline__ void slice_copy(const float* g, float* dst, int lane) {
  for (int e = lane; e < 16 * SLICE_COLS; e += 32) {
    const int r  = e >> 6;
    const int cc = e & (SLICE_COLS - 1);
    dst[r * SLICE_STRIDE + cc] = g[(size_t)r * DIMS + cc];
  }
}
#endif

// ---------------------------------------------------------------------------
// Kernel 1: squared L2 norm of every train row. One wave (32 lanes) per row.
// ---------------------------------------------------------------------------
__global__ void loop_rownorm_kernel(const float* __restrict__ P,
                                    float* __restrict__ p2, int n) {
  const int wave = threadIdx.x >> 5;
  const int lane = threadIdx.x & 31;
  const int row  = blockIdx.x * (blockDim.x >> 5) + wave;
  if (row >= n) return;
  const float* pr = P + (size_t)row * DIMS;
  float s = 0.0f;
  for (int c = lane; c < DIMS; c += 32) {
    float v = pr[c];
    s = fmaf(v, v, s);
  }
  #pragma unroll
  for (int off = 16; off > 0; off >>= 1) s += __shfl_down(s, off, 32);
  if (lane == 0) p2[row] = s;
}

// ---------------------------------------------------------------------------
// Kernel 2: TDM-staged distance GEMM (fp32 WMMA 16x16x4) + on-the-fly top-16.
// Block = 16 queries, 2 waves. Waves partition the 625 point tiles; each wave
// double-buffers 16x64 point slices through LDS via the Tensor Data Mover.
// ---------------------------------------------------------------------------
__global__ void loop_knn_kernel(const float* __restrict__ Qbase,
                                const int*   __restrict__ gather,
                                const float* __restrict__ P,
                                const float* __restrict__ p2,
                                float* __restrict__ out_pdist,
                                int*   __restrict__ out_idx,
                                int nq) {
  __shared__ float ldsQ[16 * QSTRIDE];             // queries; reused as merge space
  __shared__ float lds_q2[16];
  __shared__ float sliceA[NWAVES * 2 * SLICE_FLOATS];  // per-wave double buffers

  const int tid   = threadIdx.x;
  const int wave  = tid >> 5;
  const int lane  = tid & 31;
  const int nhalf = lane >> 4;    // K/M half select
  const int qn    = lane & 15;    // this lane's query (N) / A-row (M) index
  const int q0    = blockIdx.x * 16;

  // ---- stage 16 query rows into LDS (gathered for level 2) ----
  for (int e = tid; e < 16 * DIMS; e += BLOCKT) {
    const int r = e >> 9;           // DIMS = 512
    const int c = e & (DIMS - 1);
    const int qrow = gather ? gather[q0 + r] : (q0 + r);
    ldsQ[r * QSTRIDE + c] = Qbase[(size_t)qrow * DIMS + c];
  }
  __syncthreads();

  // ---- per-query squared norms from the staged tile ----
  if (tid < 16) {
    float s = 0.0f;
    const float* qr = &ldsQ[tid * QSTRIDE];
    for (int c = 0; c < DIMS; ++c) s = fmaf(qr[c], qr[c], s);
    lds_q2[tid] = s;
  }
  __syncthreads();

  const float q2v = lds_q2[qn];

  // ---- per-lane top-16 (clamped squared distances), lazy max tracking ----
  float best[KNN];
  int   bidx[KNN];
  #pragma unroll
  for (int j = 0; j < KNN; ++j) { best[j] = BIGF; bidx[j] = 0; }
  float worst = BIGF;

  float* wbuf = &sliceA[wave * 2 * SLICE_FLOATS];
  const float* brow  = &ldsQ[qn * QSTRIDE + 2 * nhalf];
  const float* arow0 = wbuf + qn * SLICE_STRIDE + 2 * nhalf;
  const float* arow1 = wbuf + SLICE_FLOATS + qn * SLICE_STRIDE + 2 * nhalf;
#ifdef USE_TDM
  const unsigned int ldsBase = (unsigned int)(uintptr_t)(void*)wbuf;
#endif

  const int myTiles = (NTILES - wave + NWAVES - 1) / NWAVES;
  const int nchunks = myTiles * SLICES;

  auto issue = [&](int c) {
    const int t = wave + (c >> 3) * NWAVES;
    const int s = c & (SLICES - 1);
    const float* g = P + (size_t)(t * 16) * DIMS + s * SLICE_COLS;
#ifdef USE_TDM
    tdm_load_slice(g, ldsBase + (unsigned int)((c & 1) * (SLICE_FLOATS * 4)));
#else
    slice_copy(g, wbuf + (c & 1) * SLICE_FLOATS, lane);
#endif
  };

  const v8f vzero = {};
  v8f acc = vzero;

  issue(0);
  for (int c = 0; c < nchunks; ++c) {
    const int t = wave + (c >> 3) * NWAVES;
    const int s = c & (SLICES - 1);

    if (c + 1 < nchunks) issue(c + 1);      // prefetch into other buffer
    SCHED_FENCE();
#ifdef USE_TDM
    if (c + 1 < nchunks) __builtin_amdgcn_s_wait_tensorcnt((short)1);
    else                 __builtin_amdgcn_s_wait_tensorcnt((short)0);
#endif
    SCHED_FENCE();

    const float* ar = (c & 1) ? arow1 : arow0;
    const float* br = brow + s * SLICE_COLS;
    #pragma unroll
    for (int kk = 0; kk < SLICE_COLS; kk += 4) {
      v2f a = *reinterpret_cast<const v2f*>(ar + kk);   // point fragment (LDS, TDM-staged)
      v2f b = *reinterpret_cast<const v2f*>(br + kk);   // query fragment (LDS)
      acc = __builtin_amdgcn_wmma_f32_16x16x4_f32(
          /*neg_a=*/false, a, /*neg_b=*/false, b,
          /*c_mod=*/(short)0, acc, /*reuse_a=*/false, /*reuse_b=*/false);
    }

    if (s == SLICES - 1) {                  // tile finished: fold into top-16
      const int pbase = t * 16 + 8 * nhalf;
      float p2v[8];
      #pragma unroll
      for (int r = 0; r < 8; ++r) p2v[r] = p2[pbase + r];

      #pragma unroll
      for (int r = 0; r < 8; ++r) {
        const float dot = acc[r];
        const float d2  = fmaxf(fmaf(-2.0f, dot, q2v + p2v[r]), 0.0f);
        if (d2 < worst) {
          int wi = 0; float w = best[0];
          #pragma unroll
          for (int j = 1; j < KNN; ++j) if (best[j] > w) { w = best[j]; wi = j; }
          const int pidx = pbase + r;
          #pragma unroll
          for (int j = 0; j < KNN; ++j) {
            const bool hit = (j == wi);
            best[j] = hit ? d2   : best[j];
            bidx[j] = hit ? pidx : bidx[j];
          }
          w = best[0];
          #pragma unroll
          for (int j = 1; j < KNN; ++j) w = fmaxf(w, best[j]);
          worst = w;
        }
      }
      acc = vzero;
    }
  }

  // ---- dump per-lane lists into merge space (aliases ldsQ; queries are dead) ----
  __syncthreads();
  float* cand_d = ldsQ;                         // [16][CSTRIDE] floats
  int*   cand_i = (int*)(ldsQ + 16 * CSTRIDE);  // [16][CSTRIDE] ints
  const int slot0 = (wave * 2 + nhalf) * KNN;
  #pragma unroll
  for (int j = 0; j < KNN; ++j) {
    cand_d[qn * CSTRIDE + slot0 + j] = best[j];
    cand_i[qn * CSTRIDE + slot0 + j] = bidx[j];
  }
  __syncthreads();

  if (tid < 16) {
    const int q = tid;
    float sum = 0.0f;
    for (int s = 0; s < KNN; ++s) {
      float m = cand_d[q * CSTRIDE + 0]; int mi = 0;
      for (int c = 1; c < CANDS; ++c) {
        const float v = cand_d[q * CSTRIDE + c];
        if (v < m) { m = v; mi = c; }
      }
      sum += m;                               // clamped squared distance
      if (out_idx) out_idx[(q0 + q) * KNN + s] = cand_i[q * CSTRIDE + mi];
      cand_d[q * CSTRIDE + mi] = BIGF;
    }
    out_pdist[q0 + q] = LAMBDA * sqrtf(sum * (1.0f / (float)KNN));
  }
}

// ---------------------------------------------------------------------------
// Kernel 3: lof ratio -> erf -> relu
// ---------------------------------------------------------------------------
__global__ void loop_finalize_kernel(const float* __restrict__ pdist,
                                     const float* __restrict__ pdists,
                                     float* __restrict__ out, int b) {
  const int q = blockIdx.x * blockDim.x + threadIdx.x;
  if (q >= b) return;
  float s = 0.0f;
  #pragma unroll
  for (int j = 0; j < KNN; ++j) s += pdists[q * KNN + j];
  const float nf  = s * (1.0f / (float)KNN);
  const float lof = pdist[q] / nf - 1.0f;
  const float v   = erff(lof * 0.70710678118654752440f);  // / sqrt(2), Z = 1
  out[q] = fmaxf(v, 0.0f);
}

// ---------------------------------------------------------------------------
extern "C" void kernel_launch(void* const* d_in, const int* in_sizes, int n_in,
                              void* d_out, int out_size, void* d_ws, size_t ws_size,
                              hipStream_t stream) {
  const float* X     = (const float*)d_in[0];  // [BATCH, DIMS]
  const float* train = (const float*)d_in[1];  // [N_TRAIN, DIMS]
  float* out = (float*)d_out;                  // [BATCH]

  // workspace layout
  float* p2      = (float*)d_ws;                   // [N_TRAIN]
  float* pdist1  = p2 + N_TRAIN;                   // [BATCH]
  float* pdists2 = pdist1 + BATCH;                 // [BATCH*KNN]
  int*   idx1    = (int*)(pdists2 + BATCH * KNN);  // [BATCH*KNN]

  // 1) train-row squared norms
  loop_rownorm_kernel<<<(N_TRAIN + 7) / 8, 256, 0, stream>>>(train, p2, N_TRAIN);

  // 2) level-1 kNN: queries X vs train (keeps indices)
  loop_knn_kernel<<<BATCH / 16, BLOCKT, 0, stream>>>(
      X, nullptr, train, p2, pdist1, idx1, BATCH);

  // 3) level-2 kNN: gathered neighbor points vs train (pdists only)
  loop_knn_kernel<<<(BATCH * KNN) / 16, BLOCKT, 0, stream>>>(
      train, idx1, train, p2, pdists2, nullptr, BATCH * KNN);

  // 4) finalize: lof -> erf -> relu
  loop_finalize_kernel<<<(BATCH + 255) / 256, 256, 0, stream>>>(
      pdist1, pdists2, out, BATCH);
}